// LearnableGraphBuilder_86363202387974
// MI455X (gfx1250) — compile-verified
//
#include <hip/hip_runtime.h>
#include <hip/hip_bf16.h>

// CDNA5 / gfx1250. wave32. Exact-fp32 WMMA (16x16x4) edge scores,
// async global->LDS row staging (ASYNCcnt path), NT streaming stores.

typedef __attribute__((ext_vector_type(2))) float v2f;
typedef __attribute__((ext_vector_type(4))) float v4f;
typedef __attribute__((ext_vector_type(8))) float v8f;
typedef __attribute__((ext_vector_type(4))) int   v4i;

#define N_NODES 100000
#define TOP_K   20
#define EMB_DIM 128
#define WAVES_PER_BLOCK 4
#define LANES 32

#if defined(__HIP_DEVICE_COMPILE__) && __has_builtin(__builtin_amdgcn_global_load_async_to_lds_b128)
#define HAS_ASYNC_LDS 1
typedef __attribute__((address_space(1))) v4i* gp_v4i;   // global int4*
typedef __attribute__((address_space(3))) v4i* lp_v4i;   // LDS int4*
#else
#define HAS_ASYNC_LDS 0
#endif

#if defined(__HIP_DEVICE_COMPILE__) && __has_builtin(__builtin_amdgcn_s_wait_asynccnt)
#define WAIT_ASYNC0() __builtin_amdgcn_s_wait_asynccnt(0)
#elif defined(__HIP_DEVICE_COMPILE__)
#define WAIT_ASYNC0() asm volatile("s_wait_asynccnt 0x0" ::: "memory")
#else
#define WAIT_ASYNC0()
#endif

__global__ __launch_bounds__(WAVES_PER_BLOCK * LANES)
void lgb_edges_kernel(const float* __restrict__ emb,
                      const long long* __restrict__ nbr_idx,
                      long long* __restrict__ edge_index,   // [2, N*K] int64
                      float* __restrict__ edge_weight)      // [N*K] fp32
{
    // Per-wave LDS: 20 neighbor rows + 1 own row, each EMB_DIM floats.
    __shared__ float lds[WAVES_PER_BLOCK][(TOP_K + 1) * EMB_DIM];
    __shared__ float sc[WAVES_PER_BLOCK][32];

    const int lane = threadIdx.x & 31;
    const int wid  = threadIdx.x >> 5;
    const int node = blockIdx.x * WAVES_PER_BLOCK + wid;
    if (node >= N_NODES) return;   // wave-uniform

    float* ldsW = &lds[wid][0];

    // ---- Neighbor index for this lane (lanes 0..19); NT streaming read ----
    int my_nbr = 0;
    if (lane < TOP_K)
        my_nbr = (int)__builtin_nontemporal_load(nbr_idx + (size_t)node * TOP_K + lane);

    // ---- Stage 20 neighbor rows + own row into LDS (coalesced 512B rows) ----
#if HAS_ASYNC_LDS
    // Own embedding -> LDS row TOP_K (async; each lane moves 16B)
    __builtin_amdgcn_global_load_async_to_lds_b128(
        (gp_v4i)(emb + (size_t)node * EMB_DIM + lane * 4),
        (lp_v4i)(ldsW + TOP_K * EMB_DIM + lane * 4), 0, 0);
    #pragma unroll
    for (int r = 0; r < TOP_K; ++r) {
        int row = __shfl(my_nbr, r, 32);                 // const lane -> v_readlane
        __builtin_amdgcn_global_load_async_to_lds_b128(
            (gp_v4i)(emb + (size_t)row * EMB_DIM + lane * 4),
            (lp_v4i)(ldsW + r * EMB_DIM + lane * 4), 0, 0);
    }
    WAIT_ASYNC0();                                       // ASYNCcnt -> 0
#else
    {
        v4f v = *(const v4f*)(emb + (size_t)node * EMB_DIM + lane * 4);
        *(v4f*)(ldsW + TOP_K * EMB_DIM + lane * 4) = v;
    }
    #pragma unroll
    for (int r = 0; r < TOP_K; ++r) {
        int row = __shfl(my_nbr, r, 32);                 // const lane -> v_readlane
        v4f v = *(const v4f*)(emb + (size_t)row * EMB_DIM + lane * 4);
        *(v4f*)(ldsW + r * EMB_DIM + lane * 4) = v;
    }
#endif
    // All LDS traffic is wave-private; DS ops are in-order within a wave.

    // ---- WMMA fp32 16x16x4: scores[k] = dot(nbr_k, own), exact fp32 ----
    // A(16x4): lanes 0-15 hold {K0,K1} of row M=lane; lanes 16-31 hold {K2,K3}.
    // B(4x16): all 16 columns = own-embedding chunk (broadcast).
    const int t       = lane & 15;
    const int rowA0   = t;                               // neighbors 0..15
    const int rowA1   = TOP_K - 4 + (t < 4 ? t : 3);     // neighbors 16..19 (clamped)
    const int colOff  = (lane >= 16) ? 2 : 0;            // K2/K3 half

    v8f acc0 = {};
    v8f acc1 = {};

    #pragma unroll 8
    for (int kc = 0; kc < EMB_DIM / 4; ++kc) {
        const int base = kc * 4 + colOff;
        v2f a0 = *(const v2f*)(ldsW + rowA0 * EMB_DIM + base);
        v2f a1 = *(const v2f*)(ldsW + rowA1 * EMB_DIM + base);
        v2f b  = *(const v2f*)(ldsW + TOP_K * EMB_DIM + base);
        acc0 = __builtin_amdgcn_wmma_f32_16x16x4_f32(false, a0, false, b,
                                                     (short)0, acc0, false, false);
        acc1 = __builtin_amdgcn_wmma_f32_16x16x4_f32(false, a1, false, b,
                                                     (short)0, acc1, false, false);
    }

    // ---- Extract column 0 of D: lane0 VGPR r -> M=r; lane16 VGPR r -> M=8+r ----
    if (lane == 0) {
        sc[wid][0] = acc0[0]; sc[wid][1] = acc0[1];
        sc[wid][2] = acc0[2]; sc[wid][3] = acc0[3];
        sc[wid][4] = acc0[4]; sc[wid][5] = acc0[5];
        sc[wid][6] = acc0[6]; sc[wid][7] = acc0[7];
        sc[wid][16] = acc1[0]; sc[wid][17] = acc1[1];
        sc[wid][18] = acc1[2]; sc[wid][19] = acc1[3];
    }
    if (lane == 16) {
        sc[wid][8]  = acc0[0]; sc[wid][9]  = acc0[1];
        sc[wid][10] = acc0[2]; sc[wid][11] = acc0[3];
        sc[wid][12] = acc0[4]; sc[wid][13] = acc0[5];
        sc[wid][14] = acc0[6]; sc[wid][15] = acc0[7];
    }
    // same-wave DS store->load: in-order; compiler inserts s_wait_dscnt

    // ---- Softmax over 20 scores (wave32 shuffle reduction) ----
    float s = (lane < TOP_K) ? sc[wid][lane] : -3.402823466e+38f;
    float m = s;
    #pragma unroll
    for (int off = 16; off > 0; off >>= 1)
        m = fmaxf(m, __shfl_xor(m, off, 32));
    float e = (lane < TOP_K) ? __expf(s - m) : 0.0f;
    float sum = e;
    #pragma unroll
    for (int off = 16; off > 0; off >>= 1)
        sum += __shfl_xor(sum, off, 32);
    float w = e / sum;

    // ---- Emit edges (pure streaming output: non-temporal, spare the L2) ----
    if (lane < TOP_K) {
        const size_t eoff = (size_t)node * TOP_K + lane;
        const size_t NK   = (size_t)N_NODES * TOP_K;
        __builtin_nontemporal_store((long long)node,   edge_index + eoff);      // src
        __builtin_nontemporal_store((long long)my_nbr, edge_index + NK + eoff); // dst
        __builtin_nontemporal_store(w,                 edge_weight + eoff);
    }
}

extern "C" void kernel_launch(void* const* d_in, const int* in_sizes, int n_in,
                              void* d_out, int out_size, void* d_ws, size_t ws_size,
                              hipStream_t stream) {
    const float*     emb = (const float*)d_in[0];
    const long long* idx = (const long long*)d_in[1];

    // d_out layout: edge_index [2, N*K] int64, then edge_weight [N*K] fp32
    long long* edge_index  = (long long*)d_out;
    float*     edge_weight = (float*)((char*)d_out +
                               (size_t)2 * N_NODES * TOP_K * sizeof(long long));

    const int blocks = (N_NODES + WAVES_PER_BLOCK - 1) / WAVES_PER_BLOCK; // 25000
    lgb_edges_kernel<<<blocks, WAVES_PER_BLOCK * LANES, 0, stream>>>(
        emb, idx, edge_index, edge_weight);
}